// Linear_65678639890809
// MI455X (gfx1250) — compile-verified
//
#include <hip/hip_runtime.h>

typedef float v2f __attribute__((ext_vector_type(2)));
typedef float v4f __attribute__((ext_vector_type(4)));
typedef float v8f __attribute__((ext_vector_type(8)));

#define HDIM 4096
#define ODIM 4096
#define RANK 64
#define ROWS 16384          // B*S
#define ROWS_PER_ADPT 2048  // (B/N)*S
#define KC 128              // K-chunk staged in LDS (stage 1)

// ---------------------------------------------------------------------------
// Stage 1: t[m, r] = scaling[n] * sum_h (data[m,h]*mask[m,h]) * lora_a[n, r, h]
// Grid: ROWS/16 workgroups of 128 threads (4 waves). Wave w owns ranks
// [16w, 16w+16). x-chunk (16 x KC) staged in LDS with stride KC+4 so the
// WMMA A-fragment gather (row = lane%16, k += 2*(lane>=16)) is conflict-free.
// ---------------------------------------------------------------------------
__global__ __launch_bounds__(128) void lora_stage1(
    const float* __restrict__ data, const float* __restrict__ mask,
    const float* __restrict__ lora_a, const float* __restrict__ scalings,
    float* __restrict__ t_out)
{
    __shared__ float xs[16 * (KC + 4)];

    const int m0      = blockIdx.x * 16;
    const int adapter = m0 / ROWS_PER_ADPT;
    const int tid     = threadIdx.x;
    const int wave    = tid >> 5;
    const int lane    = tid & 31;
    const int lrow    = lane & 15;          // A-fragment row (M)
    const int khalf   = (lane >> 4) << 1;   // 0 (lanes 0-15) or 2 (lanes 16-31)
    const int r0      = wave << 4;

    // B-fragment source: column n of B == row (r0+lrow) of lora_a[adapter]
    const float* __restrict__ brow =
        lora_a + ((size_t)adapter * RANK + (r0 + lrow)) * HDIM;

    v8f acc = {};

    for (int k0 = 0; k0 < HDIM; k0 += KC) {
        // Stream 16 x KC chunk of data*mask into LDS (NT 128-bit loads).
        #pragma unroll
        for (int i = 0; i < 4; ++i) {
            int f   = tid + i * 128;        // v4f index, 32 v4f per row
            int row = f >> 5;
            int cv  = f & 31;
            const v4f* dp = (const v4f*)(data + (size_t)(m0 + row) * HDIM + k0) + cv;
            const v4f* mp = (const v4f*)(mask + (size_t)(m0 + row) * HDIM + k0) + cv;
            v4f d  = __builtin_nontemporal_load(dp);
            v4f mm = __builtin_nontemporal_load(mp);
            d *= mm;
            *(v4f*)(&xs[row * (KC + 4) + (cv << 2)]) = d;
        }
        __syncthreads();

        const float* xrow = &xs[lrow * (KC + 4)];
        #pragma unroll
        for (int kk = 0; kk < KC; kk += 4) {
            v2f a = *(const v2f*)(xrow + kk + khalf);
            v2f b = *(const v2f*)(brow + k0 + kk + khalf);
            acc = __builtin_amdgcn_wmma_f32_16x16x4_f32(
                false, a, false, b, (short)0, acc, false, false);
        }
        __syncthreads();
    }

    // D-layout store: VGPR v -> row m0 + v + 8*(lane>=16), col r0 + lane%16
    const float sc     = scalings[adapter];
    const int   rowoff = (lane >> 4) << 3;
    #pragma unroll
    for (int v = 0; v < 8; ++v)
        t_out[(size_t)(m0 + rowoff + v) * RANK + r0 + lrow] = acc[v] * sc;
}

// ---------------------------------------------------------------------------
// Stage 2: out[m, o] = result[m, o] + sum_r t[m, r] * lora_b[n, o, r]
// Grid: (ROWS/16, ODIM/64) WGs of 128 threads; wave w owns cols
// [64*by + 16w, +16). result tile is loaded straight into the C accumulator.
// ---------------------------------------------------------------------------
__global__ __launch_bounds__(128) void lora_stage2(
    const float* __restrict__ t_in, const float* __restrict__ lora_b,
    const float* __restrict__ result, float* __restrict__ out)
{
    __shared__ float ts[16 * (RANK + 4)];

    const int m0      = blockIdx.x * 16;
    const int adapter = m0 / ROWS_PER_ADPT;
    const int tid     = threadIdx.x;
    const int wave    = tid >> 5;
    const int lane    = tid & 31;
    const int lrow    = lane & 15;
    const int khalf   = (lane >> 4) << 1;
    const int col     = blockIdx.y * 64 + (wave << 4) + lrow;

    // Stage the 16 x 64 t-tile into LDS (256 v4f, 2 per thread).
    #pragma unroll
    for (int i = 0; i < 2; ++i) {
        int f   = tid + i * 128;
        int row = f >> 4;                   // 16 v4f per row
        int cv  = f & 15;
        v4f d = *((const v4f*)(t_in + (size_t)(m0 + row) * RANK) + cv);
        *(v4f*)(&ts[row * (RANK + 4) + (cv << 2)]) = d;
    }
    __syncthreads();

    // C accumulator := result tile (D/C VGPR layout).
    const int rowoff = (lane >> 4) << 3;
    v8f acc;
    #pragma unroll
    for (int v = 0; v < 8; ++v)
        acc[v] = __builtin_nontemporal_load(
            result + (size_t)(m0 + rowoff + v) * ODIM + col);

    // B-fragment source: column o of B == row o of lora_b[adapter] (length R)
    const float* __restrict__ bcol =
        lora_b + ((size_t)adapter * ODIM + col) * RANK;
    const float* xrow = &ts[lrow * (RANK + 4)];

    #pragma unroll
    for (int kk = 0; kk < RANK; kk += 4) {
        v2f a = *(const v2f*)(xrow + kk + khalf);
        v2f b = *(const v2f*)(bcol + kk + khalf);
        acc = __builtin_amdgcn_wmma_f32_16x16x4_f32(
            false, a, false, b, (short)0, acc, false, false);
    }

    #pragma unroll
    for (int v = 0; v < 8; ++v)
        __builtin_nontemporal_store(
            acc[v], out + (size_t)(m0 + rowoff + v) * ODIM + col);
}

// ---------------------------------------------------------------------------
extern "C" void kernel_launch(void* const* d_in, const int* in_sizes, int n_in,
                              void* d_out, int out_size, void* d_ws, size_t ws_size,
                              hipStream_t stream) {
    const float* result   = (const float*)d_in[0];
    const float* data     = (const float*)d_in[1];
    const float* mask     = (const float*)d_in[2];
    const float* lora_a   = (const float*)d_in[3];
    const float* lora_b   = (const float*)d_in[4];
    const float* scalings = (const float*)d_in[5];
    float*       out      = (float*)d_out;
    float*       t_ws     = (float*)d_ws;   // ROWS*RANK floats = 4 MB

    (void)in_sizes; (void)n_in; (void)out_size; (void)ws_size;

    lora_stage1<<<dim3(ROWS / 16), dim3(128), 0, stream>>>(
        data, mask, lora_a, scalings, t_ws);
    lora_stage2<<<dim3(ROWS / 16, ODIM / 64), dim3(128), 0, stream>>>(
        t_ws, lora_b, result, out);
}